// PatchTSTFourierApproximator_64390149702226
// MI455X (gfx1250) — compile-verified
//
#include <hip/hip_runtime.h>

// FFT-top-k-iFFT for (B,S,C)=(64,4096,128), k read from device.
// Four-step radix-64 FFT along S, expressed as 64x64x128 f32 WMMA matmuls
// with C as the WMMA N dimension (coalesced 512B rows everywhere).
// B-operand staging uses gfx1250 GLOBAL_LOAD_ASYNC_TO_LDS_B128 (ASYNCcnt).
//
// Workspace layout (d_ws): wr[PLANE] | wi[PLANE] | thr[B*C]
//   PLANE = B*64*64*128 floats = 128MB each plane  (total ~268.5 MB)

typedef __attribute__((ext_vector_type(2))) float v2f;
typedef __attribute__((ext_vector_type(8))) float v8f;
typedef __attribute__((ext_vector_type(4))) int v4i;

#define TWO_PI 6.28318530717958647692f

constexpr int    Bn    = 64;
constexpr int    Sn    = 4096;
constexpr int    Cn    = 128;
constexpr int    Rr    = 64;                         // radix
constexpr size_t PLANE = (size_t)Bn * Rr * Rr * Cn;  // 33,554,432 floats

// ------------------------------------------------- async global->LDS staging
#if defined(__has_builtin)
#if __has_builtin(__builtin_amdgcn_global_load_async_to_lds_b128)
#define HAS_ASYNC_LDS 1
#endif
#endif
#ifndef HAS_ASYNC_LDS
#define HAS_ASYNC_LDS 0
#endif

#if HAS_ASYNC_LDS
typedef __attribute__((address_space(1))) v4i* gptr_v4i;  // global (__device__)
typedef __attribute__((address_space(3))) v4i* lptr_v4i;  // LDS (__shared__)
#endif

// Copy 16 bytes global -> LDS. Async path writes LDS directly (no VGPR data),
// tracked by ASYNCcnt.
__device__ __forceinline__ void g2l_b128(float* lds_dst, const float* gsrc) {
#if HAS_ASYNC_LDS
  __builtin_amdgcn_global_load_async_to_lds_b128(
      (gptr_v4i)const_cast<float*>(gsrc), (lptr_v4i)lds_dst,
      /*offset=*/0, /*cpol=*/0);
#else
  lds_dst[0] = gsrc[0];
  lds_dst[1] = gsrc[1];
  lds_dst[2] = gsrc[2];
  lds_dst[3] = gsrc[3];
#endif
}

__device__ __forceinline__ void wait_async_lds() {
#if HAS_ASYNC_LDS
#if __has_builtin(__builtin_amdgcn_s_wait_asynccnt)
  __builtin_amdgcn_s_wait_asynccnt(0);
#else
  asm volatile("s_wait_asynccnt 0" ::: "memory");
#endif
#endif
}

// ---------------------------------------------------------------- WMMA core
__device__ __forceinline__ v8f wmma_f32(v2f a, v2f b, v8f c) {
  // D(16x16,f32) += A(16x4,f32) * B(4x16,f32)
  return __builtin_amdgcn_wmma_f32_16x16x4_f32(
      /*neg_a=*/false, a, /*neg_b=*/false, b,
      /*c_mod=*/(short)0, c, /*reuse_a=*/false, /*reuse_b=*/false);
}

// One workgroup (256 thr = 8 waves) computes D[64][128] = A[64][64] * B[64][128]
// (complex coefficient matrix A, real or complex B, optionally imag output).
// Wave w: M-tile = w>>1 (rows m0..m0+15), N-tiles (w&1)*4 .. +3.
template <bool BCPLX, bool WIM>
__device__ __forceinline__ void mm64x64x128(const float* lAre, const float* lAim,
                                            const float* lBre, const float* lBim,
                                            v8f accRe[4], v8f accIm[4]) {
  const int t = threadIdx.x, lane = t & 31, w = t >> 5;
  const int lo = lane & 15, hi = lane >> 4;
  const int m0 = (w >> 1) << 4, ntb = (w & 1) * 4;
  for (int kk = 0; kk < 16; ++kk) {
    const int kb = (kk << 2) + (hi << 1);  // K pair base for this lane-half
    // A fragment per ISA layout: lane m = m0+lo, VGPRs = {A[m][kb], A[m][kb+1]}
    const v2f ar = *(const v2f*)&lAre[(m0 + lo) * 64 + kb];
    const v2f ai = *(const v2f*)&lAim[(m0 + lo) * 64 + kb];
    v2f ain; ain.x = -ai.x; ain.y = -ai.y;  // for Re -= Ai*Bi (f32 WMMA has no A-neg)
#pragma unroll
    for (int nt = 0; nt < 4; ++nt) {
      const int n = ((ntb + nt) << 4) + lo;
      v2f br; br.x = lBre[kb * 128 + n]; br.y = lBre[(kb + 1) * 128 + n];
      accRe[nt] = wmma_f32(ar, br, accRe[nt]);
      if (WIM) accIm[nt] = wmma_f32(ai, br, accIm[nt]);
      if (BCPLX) {
        v2f bi; bi.x = lBim[kb * 128 + n]; bi.y = lBim[(kb + 1) * 128 + n];
        accRe[nt] = wmma_f32(ain, bi, accRe[nt]);
        if (WIM) accIm[nt] = wmma_f32(ar, bi, accIm[nt]);
      }
    }
  }
}

// Fill 64x64 DFT matrix E64(sign * m*k):  lA[m*64+k] = exp(sign*2πi*m*k/64)
__device__ __forceinline__ void fill_F64(float* lAre, float* lAim, float sign) {
  for (int i = threadIdx.x; i < 4096; i += 256) {
    const int m = i >> 6, k = i & 63;
    const float ang = sign * (TWO_PI / 64.0f) * (float)((m * k) & 63);
    float s, c;
    __sincosf(ang, &s, &c);
    lAre[i] = c;
    lAim[i] = s;
  }
}

// ------------------------------------------------------------------ K1: fwd stage 1
// wg=(b,n2): Y[k1][c] = E4096(-k1*n2) * sum_n1 E64(-k1*n1) x[64*n1+n2][c]
__global__ void k1_fwd1(const float* __restrict__ x, float* __restrict__ wr,
                        float* __restrict__ wi) {
  extern __shared__ float sm[];
  float* lAre = sm;
  float* lAim = sm + 4096;
  float* lBre = sm + 8192;
  const int b = blockIdx.x >> 6, n2 = blockIdx.x & 63;
  const int t = threadIdx.x;
  const size_t xb = (size_t)b * Sn * Cn;
  // async-stage B rows (n1-th row: 512B contiguous at 32KB stride), 16B/op
  for (int i = t; i < 2048; i += 256) {
    const int n1 = i >> 5, q = (i & 31) << 2;  // q = float offset within row
    g2l_b128(&lBre[n1 * 128 + q], &x[xb + (size_t)(64 * n1 + n2) * Cn + q]);
  }
  fill_F64(lAre, lAim, -1.0f);  // overlaps with async staging
  wait_async_lds();
  __syncthreads();
  v8f z = {};
  v8f accRe[4] = {z, z, z, z}, accIm[4] = {z, z, z, z};
  mm64x64x128<false, true>(lAre, lAim, lBre, nullptr, accRe, accIm);
  const int lane = t & 31, w = t >> 5, lo = lane & 15, hi = lane >> 4;
  const int m0 = (w >> 1) << 4, ntb = (w & 1) * 4;
  const size_t ob = (size_t)b * (Rr * Rr * Cn);
#pragma unroll
  for (int nt = 0; nt < 4; ++nt) {
    const int c = ((ntb + nt) << 4) + lo;
#pragma unroll
    for (int r = 0; r < 8; ++r) {
      const int k1i = m0 + r + (hi << 3);
      float sn, cs;
      __sincosf(-(TWO_PI / 4096.0f) * (float)((k1i * n2) & 4095), &sn, &cs);
      const float dr = accRe[nt][r], di = accIm[nt][r];
      const size_t idx = ob + ((size_t)k1i * 64 + n2) * 128 + c;
      wr[idx] = dr * cs - di * sn;
      wi[idx] = dr * sn + di * cs;
    }
  }
}

// ------------------------------------------------------------------ K2: fwd stage 2
// wg=(b,k1): S[k1][k2][c] = sum_n2 E64(-k2*n2) Y[k1][n2][c]   (in-place block)
__global__ void k2_fwd2(float* __restrict__ wr, float* __restrict__ wi) {
  extern __shared__ float sm[];
  float* lAre = sm;
  float* lAim = sm + 4096;
  float* lBre = sm + 8192;
  float* lBim = sm + 16384;
  const int b = blockIdx.x >> 6, k1 = blockIdx.x & 63;
  const int t = threadIdx.x;
  const size_t base = (size_t)b * (Rr * Rr * Cn) + (size_t)k1 * (Rr * Cn);
  for (int i = t; i < 2048; i += 256) {  // contiguous 32KB block per plane
    const int q = i << 2;
    g2l_b128(&lBre[q], &wr[base + q]);
    g2l_b128(&lBim[q], &wi[base + q]);
  }
  fill_F64(lAre, lAim, -1.0f);
  wait_async_lds();
  __syncthreads();
  v8f z = {};
  v8f accRe[4] = {z, z, z, z}, accIm[4] = {z, z, z, z};
  mm64x64x128<true, true>(lAre, lAim, lBre, lBim, accRe, accIm);
  const int lane = t & 31, w = t >> 5, lo = lane & 15, hi = lane >> 4;
  const int m0 = (w >> 1) << 4, ntb = (w & 1) * 4;
#pragma unroll
  for (int nt = 0; nt < 4; ++nt) {
    const int c = ((ntb + nt) << 4) + lo;
#pragma unroll
    for (int r = 0; r < 8; ++r) {
      const int k2i = m0 + r + (hi << 3);
      const size_t idx = base + (size_t)k2i * 128 + c;
      wr[idx] = accRe[nt][r];
      wi[idx] = accIm[nt][r];
    }
  }
}

// ------------------------------------------------------------------ K3: top-k threshold
// wg=(b,c): threshold = k-th largest |S[k]|^2 over rfft bins k=0..2048
__global__ void k3_topk(const float* __restrict__ wr, const float* __restrict__ wi,
                        const int* __restrict__ kptr, float* __restrict__ thr) {
  __shared__ float m2[2049];
  __shared__ float rv[256];
  __shared__ int ri[256];
  const int b = blockIdx.x >> 7, c = blockIdx.x & 127;
  const int t = threadIdx.x;
  const size_t base = (size_t)b * (Rr * Rr * Cn);
  for (int k = t; k <= 2048; k += 256) {
    const int k1 = k & 63, k2 = k >> 6;  // bin k = k1 + 64*k2
    const size_t idx = base + ((size_t)k1 * 64 + k2) * 128 + c;
    const float re = wr[idx], im = wi[idx];
    m2[k] = re * re + im * im;
  }
  __syncthreads();
  int rounds = kptr[0];
  if (rounds > 2049) rounds = 2049;
  float kth = 3.4e38f;
  for (int rd = 0; rd < rounds; ++rd) {
    float bv = -1.0f;
    int bi = -1;
    for (int k = t; k <= 2048; k += 256)
      if (m2[k] > bv) { bv = m2[k]; bi = k; }
    rv[t] = bv;
    ri[t] = bi;
    __syncthreads();
    for (int off = 128; off > 0; off >>= 1) {
      if (t < off && rv[t + off] > rv[t]) { rv[t] = rv[t + off]; ri[t] = ri[t + off]; }
      __syncthreads();
    }
    kth = rv[0];
    if (t == 0 && ri[0] >= 0) m2[ri[0]] = -2.0f;
    __syncthreads();
  }
  if (t == 0) thr[b * 128 + c] = kth;
}

// ------------------------------------------------------------------ K4: inv stage 1
// wg=(b,k1): V[k1][s1][c] = E4096(+k1*s1) * sum_k2 E64(+s1*k2) * mask(S[k1][k2][c])
__global__ void k4_inv1(float* __restrict__ wr, float* __restrict__ wi,
                        const float* __restrict__ thr) {
  extern __shared__ float sm[];
  float* lAre = sm;
  float* lAim = sm + 4096;
  float* lBre = sm + 8192;
  float* lBim = sm + 16384;
  float* lthr = sm + 24576;
  const int b = blockIdx.x >> 6, k1 = blockIdx.x & 63;
  const int t = threadIdx.x;
  const size_t base = (size_t)b * (Rr * Rr * Cn) + (size_t)k1 * (Rr * Cn);
  for (int i = t; i < 2048; i += 256) {  // async raw copy, mask applied in LDS below
    const int q = i << 2;
    g2l_b128(&lBre[q], &wr[base + q]);
    g2l_b128(&lBim[q], &wi[base + q]);
  }
  fill_F64(lAre, lAim, +1.0f);
  if (t < 128) lthr[t] = thr[b * 128 + t];
  wait_async_lds();
  __syncthreads();
  // Hermitian-symmetric mask, in place in LDS
  for (int i = t; i < 8192; i += 256) {
    const int c = i & 127;
    const float re = lBre[i], im = lBim[i];
    const bool keep = (re * re + im * im) >= lthr[c];
    lBre[i] = keep ? re : 0.0f;
    lBim[i] = keep ? im : 0.0f;
  }
  __syncthreads();
  v8f z = {};
  v8f accRe[4] = {z, z, z, z}, accIm[4] = {z, z, z, z};
  mm64x64x128<true, true>(lAre, lAim, lBre, lBim, accRe, accIm);
  const int lane = t & 31, w = t >> 5, lo = lane & 15, hi = lane >> 4;
  const int m0 = (w >> 1) << 4, ntb = (w & 1) * 4;
#pragma unroll
  for (int nt = 0; nt < 4; ++nt) {
    const int c = ((ntb + nt) << 4) + lo;
#pragma unroll
    for (int r = 0; r < 8; ++r) {
      const int s1i = m0 + r + (hi << 3);
      float sn, cs;
      __sincosf((TWO_PI / 4096.0f) * (float)((k1 * s1i) & 4095), &sn, &cs);
      const float dr = accRe[nt][r], di = accIm[nt][r];
      const size_t idx = base + (size_t)s1i * 128 + c;
      wr[idx] = dr * cs - di * sn;
      wi[idx] = dr * sn + di * cs;
    }
  }
}

// ------------------------------------------------------------------ K5: inv stage 2
// wg=(b,s1): out[s1+64*s2][c] = (1/4096) * Re( sum_k1 E64(+s2*k1) V[k1][s1][c] )
__global__ void k5_inv2(const float* __restrict__ wr, const float* __restrict__ wi,
                        float* __restrict__ out) {
  extern __shared__ float sm[];
  float* lAre = sm;
  float* lAim = sm + 4096;
  float* lBre = sm + 8192;
  float* lBim = sm + 16384;
  const int b = blockIdx.x >> 6, s1 = blockIdx.x & 63;
  const int t = threadIdx.x;
  const size_t bb = (size_t)b * (Rr * Rr * Cn);
  for (int i = t; i < 2048; i += 256) {  // row k1 (stride 32KB), 512B rows, 16B/op
    const int k1 = i >> 5, q = (i & 31) << 2;
    const size_t idx = bb + (size_t)k1 * (Rr * Cn) + (size_t)s1 * 128 + q;
    g2l_b128(&lBre[k1 * 128 + q], &wr[idx]);
    g2l_b128(&lBim[k1 * 128 + q], &wi[idx]);
  }
  fill_F64(lAre, lAim, +1.0f);
  wait_async_lds();
  __syncthreads();
  v8f z = {};
  v8f accRe[4] = {z, z, z, z}, accIm[4] = {z, z, z, z};
  mm64x64x128<true, false>(lAre, lAim, lBre, lBim, accRe, accIm);
  const int lane = t & 31, w = t >> 5, lo = lane & 15, hi = lane >> 4;
  const int m0 = (w >> 1) << 4, ntb = (w & 1) * 4;
#pragma unroll
  for (int nt = 0; nt < 4; ++nt) {
    const int c = ((ntb + nt) << 4) + lo;
#pragma unroll
    for (int r = 0; r < 8; ++r) {
      const int s2i = m0 + r + (hi << 3);
      out[((size_t)(b * Sn + s1 + 64 * s2i)) * Cn + c] = accRe[nt][r] * (1.0f / 4096.0f);
    }
  }
}

// ------------------------------------------------------------------ launch
extern "C" void kernel_launch(void* const* d_in, const int* in_sizes, int n_in,
                              void* d_out, int out_size, void* d_ws, size_t ws_size,
                              hipStream_t stream) {
  (void)in_sizes; (void)n_in; (void)out_size; (void)ws_size;
  const float* x = (const float*)d_in[0];
  const int* kptr = (const int*)d_in[1];
  float* out = (float*)d_out;
  float* wr = (float*)d_ws;
  float* wi = wr + PLANE;
  float* thr = wi + PLANE;

  const size_t smem = (size_t)(4096 * 2 + 8192 * 2 + 128) * sizeof(float);  // ~96.5 KB
  const dim3 blk(256);
  k1_fwd1<<<dim3(Bn * 64), blk, smem, stream>>>(x, wr, wi);
  k2_fwd2<<<dim3(Bn * 64), blk, smem, stream>>>(wr, wi);
  k3_topk<<<dim3(Bn * Cn), blk, 0, stream>>>(wr, wi, kptr, thr);
  k4_inv1<<<dim3(Bn * 64), blk, smem, stream>>>(wr, wi, thr);
  k5_inv2<<<dim3(Bn * 64), blk, smem, stream>>>(wr, wi, out);
}